// FNN_82549271429398
// MI455X (gfx1250) — compile-verified
//
#include <hip/hip_runtime.h>
#include <hip/hip_bf16.h>

typedef float v2f __attribute__((ext_vector_type(2)));
typedef float v8f __attribute__((ext_vector_type(8)));

// Problem constants (match reference setup)
constexpr int Bn = 128;   // batch
constexpr int Tn = 50;    // steps
constexpr int Kn = 100;   // KC rows
constexpr int Mn = 8;     // fuzzy terms
constexpr int Cn = 32;    // cognition dims
constexpr int Rn = Mn * Cn;  // 256 = rule length

// ---------------------------------------------------------------------------
// mem[b,k,c] = cognition_0[k,c]  (broadcast init each call -> deterministic)
// ---------------------------------------------------------------------------
__global__ void fnn_init_mem(const float* __restrict__ cog0, float* __restrict__ mem) {
    int i = blockIdx.x * blockDim.x + threadIdx.x;
    if (i < Bn * Kn * Cn) mem[i] = cog0[i % (Kn * Cn)];
}

// ---------------------------------------------------------------------------
// One block of 256 threads = 8 wave32s. Wave w owns batch rows [16w, 16w+16).
// Per time step: each wave does [16 x 256] x [256 x 32] via 128x
// v_wmma_f32_16x16x4_f32 (2 N-tiles x 64 k-steps), then sigmoid/normalize,
// scatter-update mem, and predict.
// ---------------------------------------------------------------------------
__global__ __launch_bounds__(256) void fnn_scan_kernel(
    const float* __restrict__ scores,   // [B, T+1]
    const int*   __restrict__ skills,   // [B, T+1]
    const float* __restrict__ mean,     // [M]
    const float* __restrict__ sigma,    // [M]
    const float* __restrict__ Wcog,     // [C, M*C]
    const float* __restrict__ Wpred,    // [C]
    const float* __restrict__ bpred,    // [1]
    float*       __restrict__ mem,      // [B, K, C] persistent state (d_ws)
    float*       __restrict__ out)      // p_t | p_t1 | g_t | g_t1, each [B,T]
{
    // B-operand staging: Bw[nt][kk][h][n][d] = Wcog[nt*16+n][4*kk + 2*h + d]
    // matches V_WMMA B 4x16 layout: VGPR d holds K = 2*half + d, N = lane16.
    __shared__ float Bw[2][64][2][16][2];   // 32 KB
    __shared__ float ldsPt[8][16];          // per-wave p_t staging

    const int tid  = threadIdx.x;
    const int wave = tid >> 5;
    const int lane = tid & 31;
    const int half = lane >> 4;
    const int l16  = lane & 15;
    const int bbase = wave * 16;
    const int rowb  = bbase + l16;          // batch row this lane builds A for
    const int S = Tn + 1;                   // score/skill row stride

    // Stage W_cog into LDS (once)
    for (int i = tid; i < 2 * 64 * 2 * 16 * 2; i += 256) {
        int d  = i & 1;
        int n  = (i >> 1) & 15;
        int h  = (i >> 5) & 1;
        int kk = (i >> 6) & 63;
        int nt = (i >> 12) & 1;
        Bw[nt][kk][h][n][d] = Wcog[(nt * 16 + n) * Rn + 4 * kk + 2 * h + d];
    }

    float meanv[Mn], isig2[Mn];
#pragma unroll
    for (int i = 0; i < Mn; ++i) {
        meanv[i] = mean[i];
        float s = sigma[i];
        isig2[i] = 1.0f / (s * s);
    }
    const float wp0 = Wpred[l16];
    const float wp1 = Wpred[16 + l16];
    const float bp  = bpred[0];
    __syncthreads();

    for (int t = 0; t < Tn; ++t) {
        // per-row scalars
        const int   ks = skills[rowb * S + t];
        const int   k1 = skills[rowb * S + t + 1];
        const float x  = scores[rowb * S + t];
        const float x1 = scores[rowb * S + t + 1];

        // fuzzy membership fs[m] = exp(-(x-mean)^2 / sigma^2)
        float fs[Mn];
#pragma unroll
        for (int i = 0; i < Mn; ++i) {
            float d = x - meanv[i];
            fs[i] = __expf(-d * d * isig2[i]);
        }

        // load old mem row (this lane's half of the 32 cols, in WMMA-A order)
        // memv[2q+d] = mem[b, ks, 4q + 2*half + d]
        float memv[16];
        const float* mrow = mem + ((size_t)rowb * Kn + ks) * Cn;
#pragma unroll
        for (int q = 0; q < 8; ++q) {
            float2 v = *(const float2*)(mrow + 4 * q + 2 * half);
            memv[2 * q]     = v.x;
            memv[2 * q + 1] = v.y;
        }

        // [16 x 256] x [256 x 32] via 64 k-steps of 16x16x4, 2 N-tiles
        v8f acc0 = {};
        v8f acc1 = {};
#pragma unroll
        for (int kk = 0; kk < 64; ++kk) {
            const int i = kk >> 3;          // fuzzy index of this K-slice
            const int q = (kk & 7) * 2;     // mem-col pair within memv
            v2f a;
            a[0] = fs[i] * memv[q];         // A[row=l16][K = 4kk + 2*half + 0]
            a[1] = fs[i] * memv[q + 1];     // A[row=l16][K = 4kk + 2*half + 1]
            v2f b0 = *(const v2f*)&Bw[0][kk][half][l16][0];
            v2f b1 = *(const v2f*)&Bw[1][kk][half][l16][0];
            acc0 = __builtin_amdgcn_wmma_f32_16x16x4_f32(
                false, a, false, b0, (short)0, acc0, false, false);
            acc1 = __builtin_amdgcn_wmma_f32_16x16x4_f32(
                false, a, false, b1, (short)0, acc1, false, false);
        }

        // sigmoid, row-normalize, scatter-update mem, predict p_t
        float sig0[8], sig1[8];
#pragma unroll
        for (int v = 0; v < 8; ++v) {
            sig0[v] = 1.0f / (1.0f + __expf(-acc0[v]));
            sig1[v] = 1.0f / (1.0f + __expf(-acc1[v]));
        }
#pragma unroll
        for (int v = 0; v < 8; ++v) {
            const int m = v + 8 * half;           // GEMM row -> batch bbase+m
            // row sum across the 16 lanes of this half (cols) and both tiles
            float sv = sig0[v] + sig1[v];
            sv += __shfl_xor(sv, 1);
            sv += __shfl_xor(sv, 2);
            sv += __shfl_xor(sv, 4);
            sv += __shfl_xor(sv, 8);
            const float inv = 1.0f / sv;
            const float n0 = sig0[v] * inv;
            const float n1 = sig1[v] * inv;
            // scatter-update mem_new row (lane m of this wave holds ks for row m)
            const int ksm = __shfl(ks, m);
            float* mw = mem + ((size_t)(bbase + m) * Kn + ksm) * Cn;
            mw[l16]      = n0;
            mw[16 + l16] = n1;
            // p_t = sigmoid(dot(norm_row, W_pred) + b_pred)
            float dv = n0 * wp0 + n1 * wp1;
            dv += __shfl_xor(dv, 1);
            dv += __shfl_xor(dv, 2);
            dv += __shfl_xor(dv, 4);
            dv += __shfl_xor(dv, 8);
            const float p = 1.0f / (1.0f + __expf(-(dv + bp)));
            if (l16 == v) {
                ldsPt[wave][m] = p;
                out[(bbase + m) * Tn + t] = p;                 // p_t
            }
        }
        __syncthreads();  // ldsPt visible; mem scatter ordered vs. reads below

        // p_t1: row k1 of mem_new for this lane's own batch row
        float p1;
        if (k1 == ks) {
            p1 = ldsPt[wave][l16];   // same row just updated -> equals p_t
        } else {
            // untouched row: dot(mem[b,k1,:], W_pred); halves split 16+16 cols
            const float* mr1 = mem + ((size_t)rowb * Kn + k1) * Cn;
            float pd = 0.0f;
#pragma unroll
            for (int c = 0; c < 16; ++c)
                pd += mr1[half * 16 + c] * Wpred[half * 16 + c];
            pd += __shfl_xor(pd, 16);
            p1 = 1.0f / (1.0f + __expf(-(pd + bp)));
        }
        if (half == 0) {
            out[1 * Bn * Tn + rowb * Tn + t] = p1;   // p_t1
            out[2 * Bn * Tn + rowb * Tn + t] = x;    // g_t  (one-hot sum == score)
            out[3 * Bn * Tn + rowb * Tn + t] = x1;   // g_t1
        }

        __threadfence();   // make this step's mem stores visible to next step
        __syncthreads();
    }
}

extern "C" void kernel_launch(void* const* d_in, const int* in_sizes, int n_in,
                              void* d_out, int out_size, void* d_ws, size_t ws_size,
                              hipStream_t stream) {
    const float* scores = (const float*)d_in[0];   // [B, T+1]
    const int*   skills = (const int*)d_in[1];     // [B, T+1]
    // d_in[2] = T (scalar, compile-time constant here)
    const float* mean   = (const float*)d_in[3];   // [M]
    const float* sigma  = (const float*)d_in[4];   // [M]
    const float* Wcog   = (const float*)d_in[5];   // [C, M*C]
    const float* Wpred  = (const float*)d_in[6];   // [C]
    const float* bpred  = (const float*)d_in[7];   // [1]
    const float* cog0   = (const float*)d_in[8];   // [K, C]

    float* mem = (float*)d_ws;                     // [B, K, C] = 1.6 MB
    float* out = (float*)d_out;

    const int nMem = Bn * Kn * Cn;
    fnn_init_mem<<<(nMem + 255) / 256, 256, 0, stream>>>(cog0, mem);
    fnn_scan_kernel<<<1, 256, 0, stream>>>(scores, skills, mean, sigma,
                                           Wcog, Wpred, bpred, mem, out);
}